// GNN_14894946582659
// MI455X (gfx1250) — compile-verified
//
#include <hip/hip_runtime.h>
#include <hip/hip_bf16.h>
#include <stdint.h>

// ---------------------------------------------------------------------------
// GCNConv (add self-loops, symmetric norm) -> tanh -> mean pool
//   x:[N,128] f32, edge_index:[2,E] i64, W:[128,16] f32, b:[16] f32
//   out: [1,16] f32
// Strategy: h = x@W via V_WMMA_F32_16X16X4_F32 (fp32 WMMA, 16x16 tile/wave),
// then edge scatter with f32 L2 atomics (h + acc both fit in 192MB L2),
// then fused self-loop/bias/tanh/mean reduction.
// ---------------------------------------------------------------------------

#define F_IN  128
#define F_OUT 16

typedef __attribute__((ext_vector_type(2))) float v2f;
typedef __attribute__((ext_vector_type(8))) float v8f;

// ---------------- init: deg=1 (self loop), acc=0, partial=0 ----------------
__global__ void gcn_init(float* __restrict__ deg, float* __restrict__ acc,
                         float* __restrict__ partial, int N) {
  int64_t total = (int64_t)N * F_OUT;
  int64_t stride = (int64_t)gridDim.x * blockDim.x;
  for (int64_t i = (int64_t)blockIdx.x * blockDim.x + threadIdx.x; i < total; i += stride) {
    acc[i] = 0.0f;
    if (i < N) deg[i] = 1.0f;
    if (i < F_OUT) partial[i] = 0.0f;
  }
}

// ---------------- degree of target nodes ----------------
__global__ void gcn_degree(const long long* __restrict__ ei, float* __restrict__ deg,
                           int64_t E) {
  int64_t stride = (int64_t)gridDim.x * blockDim.x;
  for (int64_t e = (int64_t)blockIdx.x * blockDim.x + threadIdx.x; e < E; e += stride) {
    int col = (int)ei[e + E];
    atomicAdd(&deg[col], 1.0f);
  }
}

// ---------------- deg -> d^{-1/2} in place ----------------
__global__ void gcn_dinv(float* __restrict__ deg, int N) {
  int64_t stride = (int64_t)gridDim.x * blockDim.x;
  for (int64_t i = (int64_t)blockIdx.x * blockDim.x + threadIdx.x; i < N; i += stride) {
    float d = deg[i];
    deg[i] = (d > 0.0f) ? rsqrtf(d) : 0.0f;
  }
}

// ---------------- h = x @ W with V_WMMA_F32_16X16X4_F32 ----------------
// One wave computes a 16(M=nodes) x 16(N=features) f32 tile, K=128 in 32
// chunks of 4. A-frag layout (32b A 16x4): lanes 0-15 hold {K0,K1}, lanes
// 16-31 hold {K2,K3}, M = lane%16. B mirrors with N = lane%16. C/D: 8 VGPRs,
// VGPR r = row r (lanes 0-15) / row r+8 (lanes 16-31).
__global__ void __launch_bounds__(256)
gcn_gemm_wmma(const float* __restrict__ x, const float* __restrict__ W,
              float* __restrict__ h, int N) {
  __shared__ float Wlds[F_IN * F_OUT];  // 8 KB
  // cooperative, coalesced stage of W
  for (int i = threadIdx.x; i < (F_IN * F_OUT) / 4; i += blockDim.x) {
    ((float4*)Wlds)[i] = ((const float4*)W)[i];
  }
  __syncthreads();

  const int lane = threadIdx.x & 31;
  const int wave = threadIdx.x >> 5;
  const int tile = blockIdx.x * 8 + wave;          // 16 nodes per tile
  const int64_t base = (int64_t)tile * 16;
  if (base >= N) return;                           // wave-uniform: EXEC stays full

  const int m = lane & 15;                         // row (A) / col (B,D) index
  const int khalf = (lane >> 4) << 1;              // 0 or 2

  // preload all 32 B fragments (W is shared by every tile)
  v2f Bfrag[32];
#pragma unroll
  for (int kc = 0; kc < 32; ++kc) {
    const int kb = kc * 4 + khalf;
    Bfrag[kc].x = Wlds[kb * F_OUT + m];
    Bfrag[kc].y = Wlds[(kb + 1) * F_OUT + m];
  }

  int64_t row = base + m;
  if (row >= N) row = N - 1;                       // clamp (keeps EXEC all-ones)
  const float* __restrict__ xrow = x + row * F_IN;

  v8f c = {};                                      // zero accumulator
#pragma unroll
  for (int kc = 0; kc < 32; ++kc) {
    const int kb = kc * 4 + khalf;
    v2f a;
    a.x = xrow[kb];
    a.y = xrow[kb + 1];
    // 8 args: (neg_a, A, neg_b, B, c_mod, C, reuse_a, reuse_b)
    c = __builtin_amdgcn_wmma_f32_16x16x4_f32(false, a, false, Bfrag[kc],
                                              (short)0, c, false, false);
  }

  // store D: lane<16 -> rows base+0..7, lane>=16 -> rows base+8..15, col m
  const int moff = (lane < 16) ? 0 : 8;
#pragma unroll
  for (int r = 0; r < 8; ++r) {
    const int64_t node = base + moff + r;
    if (node < N) h[node * F_OUT + m] = c[r];
  }
}

// ---------------- edge scatter: acc[col] += h[row] * dinv[row]*dinv[col] ----
// lane -> (edge, feature): 16 lanes share one edge => 64B contiguous gather
// of h[row][:] and 16 parallel f32 atomics into L2.
__global__ void gcn_scatter(const long long* __restrict__ ei,
                            const float* __restrict__ h,
                            const float* __restrict__ dinv,
                            float* __restrict__ acc, int64_t E) {
  const int64_t total = E * F_OUT;
  const int64_t stride = (int64_t)gridDim.x * blockDim.x;
  for (int64_t idx = (int64_t)blockIdx.x * blockDim.x + threadIdx.x; idx < total;
       idx += stride) {
    const int64_t e = idx >> 4;
    const int f = (int)(idx & 15);
    const int row = (int)ei[e];        // source
    const int col = (int)ei[e + E];    // target
    const float norm = dinv[row] * dinv[col];
    atomicAdd(&acc[(int64_t)col * F_OUT + f], h[(int64_t)row * F_OUT + f] * norm);
  }
}

// ---------------- self-loop + bias + tanh + mean reduction ----------------
__global__ void gcn_finalize(const float* __restrict__ acc,
                             const float* __restrict__ h,
                             const float* __restrict__ dinv,
                             const float* __restrict__ b,
                             float* __restrict__ partial, int N) {
  __shared__ float s[F_OUT];
  if (threadIdx.x < F_OUT) s[threadIdx.x] = 0.0f;
  __syncthreads();

  const int f = threadIdx.x & 15;                  // fixed per thread (stride % 16 == 0)
  const float bf = b[f];
  float sum = 0.0f;
  const int64_t total = (int64_t)N * F_OUT;
  const int64_t stride = (int64_t)gridDim.x * blockDim.x;
  for (int64_t idx = (int64_t)blockIdx.x * blockDim.x + threadIdx.x; idx < total;
       idx += stride) {
    const int64_t i = idx >> 4;
    const float di = dinv[i];
    const float v = acc[idx] + h[idx] * di * di + bf;  // self-loop norm = 1/deg
    sum += tanhf(v);
  }
  atomicAdd(&s[f], sum);                           // ds_add_f32
  __syncthreads();
  if (threadIdx.x < F_OUT) atomicAdd(&partial[threadIdx.x], s[threadIdx.x]);
}

__global__ void gcn_writeout(const float* __restrict__ partial,
                             float* __restrict__ out, int N) {
  if (threadIdx.x < F_OUT) out[threadIdx.x] = partial[threadIdx.x] * (1.0f / (float)N);
}

// ---------------------------------------------------------------------------
extern "C" void kernel_launch(void* const* d_in, const int* in_sizes, int n_in,
                              void* d_out, int out_size, void* d_ws, size_t ws_size,
                              hipStream_t stream) {
  const float* x      = (const float*)d_in[0];
  const long long* ei = (const long long*)d_in[1];   // int64 edge_index [2,E]
  const float* W      = (const float*)d_in[2];
  const float* b      = (const float*)d_in[3];
  float* out          = (float*)d_out;

  const int     N = in_sizes[0] / F_IN;
  const int64_t E = (int64_t)in_sizes[1] / 2;

  // workspace layout (floats): deg/dinv [N] | h [16N] | acc [16N] | partial [16]
  float* deg     = (float*)d_ws;
  float* h       = deg + N;
  float* acc     = h + (int64_t)N * F_OUT;
  float* partial = acc + (int64_t)N * F_OUT;

  const int threads = 256;

  gcn_init<<<2048, threads, 0, stream>>>(deg, acc, partial, N);
  gcn_degree<<<4096, threads, 0, stream>>>(ei, deg, E);
  gcn_dinv<<<512, threads, 0, stream>>>(deg, N);

  const int tiles  = (N + 15) / 16;
  const int blocks = (tiles + 7) / 8;               // 8 waves (tiles) per block
  gcn_gemm_wmma<<<blocks, threads, 0, stream>>>(x, W, h, N);

  gcn_scatter<<<8192, threads, 0, stream>>>(ei, h, deg, acc, E);
  gcn_finalize<<<4096, threads, 0, stream>>>(acc, h, deg, b, partial, N);
  gcn_writeout<<<1, 32, 0, stream>>>(partial, out, N);
}